// SpatialGaussianKernel_61220463837612
// MI455X (gfx1250) — compile-verified
//
#include <hip/hip_runtime.h>
#include <math.h>

// Gaussian blur sigma=7, K=29, reflect pad 14, input (8,64,256,256) fp32.
// Separable conv fused in LDS; each 1D pass done as banded matmul on
// V_WMMA_F32_16X16X4_F32 (K=44 band split into 11 K=4 chunks).

typedef float v2f __attribute__((ext_vector_type(2)));
typedef float v8f __attribute__((ext_vector_type(8)));

#define IMG_H 256
#define IMG_W 256
#define TS 64          // output tile (64x64)
#define PADR 14
#define IN_ROWS 96     // 64 + 28, rounded to 6 row-tiles of 16
#define IN_LD 97       // odd stride -> conflict-free strided lane loads
#define MID_ROWS 96
#define MID_LD 82      // 8*82%64=16, 2*82%64=36 -> halves hit disjoint banks

__device__ __forceinline__ float gw(int i, float invS) {
    // 1D normalized Gaussian tap, 0 outside [0,28]
    if (i < 0 || i > 28) return 0.f;
    float d = (float)(i - 14);
    return expf(-d * d * (1.0f / 98.0f)) * invS;   // 2*sigma^2 = 98
}

__global__ __launch_bounds__(256)
void gauss_blur_wmma(const float* __restrict__ xin, float* __restrict__ out) {
    __shared__ float lds[IN_ROWS * IN_LD + MID_ROWS * MID_LD];
    float* in_s = lds;
    float* mid  = lds + IN_ROWS * IN_LD;

    const int tid  = threadIdx.x;
    // wave id is uniform across the wave: pin it to an SGPR so tile loops
    // get scalar control flow (keeps EXEC all-ones around WMMA, lets the
    // compiler fully unroll and pipeline across tiles).
    const int wvu  = __builtin_amdgcn_readfirstlane(tid >> 5);   // 0..7
    const int lane = tid & 31;
    const int h    = lane >> 4;       // lane half
    const int nn   = lane & 15;

    const int x0 = blockIdx.x * TS;
    const int y0 = blockIdx.y * TS;
    const long zb = (long)blockIdx.z * (IMG_H * IMG_W);
    const float* src = xin + zb;

    // ---- phase 1: global -> LDS, 96 rows x 92 cols with reflect padding.
    // Wave wvu owns rows [12*wvu, 12*wvu+12); lanes cover cols {L, L+32, L+64}.
#pragma unroll
    for (int j = 0; j < 12; ++j) {
        int r  = wvu * 12 + j;
        int gy = y0 - PADR + r;
        gy = gy < 0 ? -gy : gy;  gy = gy > 255 ? 510 - gy : gy;
        const float* srow = src + gy * IMG_W;
        float* drow = in_s + r * IN_LD;
#pragma unroll
        for (int cc = 0; cc < 3; ++cc) {
            int c = lane + 32 * cc;
            if (c < 92) {
                int gx = x0 - PADR + c;
                gx = gx < 0 ? -gx : gx;  gx = gx > 255 ? 510 - gx : gx;
                drow[c] = srow[gx];
            }
        }
    }

    // ---- per-lane band constants: Band[k'][n] = w[k'-n], K chunked by 4.
    // A-layout (vertical pass) and B-layout (horizontal pass) coincide:
    // lane holds (m|n)=nn, K = 4c + 2h + {0,1}.
    float S = 0.f;
#pragma unroll
    for (int j = 0; j < 29; ++j) {
        float d = (float)(j - 14);
        S += expf(-d * d * (1.0f / 98.0f));
    }
    const float invS = 1.0f / S;
    v2f band[11];
#pragma unroll
    for (int c = 0; c < 11; ++c) {
        int i0 = 4 * c + 2 * h - nn;
        band[c].x = gw(i0,     invS);
        band[c].y = gw(i0 + 1, invS);
    }

    __syncthreads();

    // ---- phase 2: horizontal pass, mid(96x64) = in_s(96x44-band) x Band ----
    // 24 tiles (6 row-tiles x 4 col-tiles), 3 per wave, constant trip count.
#pragma unroll
    for (int tt = 0; tt < 3; ++tt) {
        int t  = wvu + 8 * tt;                  // 0..23, wave-uniform
        int R  = (t >> 2) * 16;
        int X0 = (t & 3) * 16;
        const float* ar = &in_s[(R + nn) * IN_LD + X0 + 2 * h];
        v8f acc = {0.f, 0.f, 0.f, 0.f, 0.f, 0.f, 0.f, 0.f};
#pragma unroll
        for (int c = 0; c < 11; ++c) {
            v2f a;
            a.x = ar[4 * c];
            a.y = ar[4 * c + 1];
            acc = __builtin_amdgcn_wmma_f32_16x16x4_f32(
                false, a, false, band[c], (short)0, acc, false, false);
        }
        float* mp = &mid[(R + 8 * h) * MID_LD + X0 + nn];
#pragma unroll
        for (int v = 0; v < 8; ++v) mp[v * MID_LD] = acc[v];
    }

    __syncthreads();

    // ---- phase 3: vertical pass, out(64x64) = Band x mid(44x16 slabs) ----
    // 16 tiles (4x4), 2 per wave, constant trip count.
    float* dst = out + zb;
#pragma unroll
    for (int tt = 0; tt < 2; ++tt) {
        int t  = wvu + 8 * tt;                  // 0..15, wave-uniform
        int Y0 = (t >> 2) * 16;
        int X0 = (t & 3) * 16;
        const float* bp = &mid[(Y0 + 2 * h) * MID_LD + X0 + nn];
        v8f acc = {0.f, 0.f, 0.f, 0.f, 0.f, 0.f, 0.f, 0.f};
#pragma unroll
        for (int c = 0; c < 11; ++c) {
            v2f b;
            b.x = bp[(4 * c)     * MID_LD];
            b.y = bp[(4 * c + 1) * MID_LD];
            acc = __builtin_amdgcn_wmma_f32_16x16x4_f32(
                false, band[c], false, b, (short)0, acc, false, false);
        }
        float* op = dst + (long)(y0 + Y0 + 8 * h) * IMG_W + x0 + X0 + nn;
#pragma unroll
        for (int v = 0; v < 8; ++v) op[(long)v * IMG_W] = acc[v];
    }
}

extern "C" void kernel_launch(void* const* d_in, const int* in_sizes, int n_in,
                              void* d_out, int out_size, void* d_ws, size_t ws_size,
                              hipStream_t stream) {
    const float* x = (const float*)d_in[0];
    float* out = (float*)d_out;
    dim3 grid(IMG_W / TS, IMG_H / TS, 8 * 64);   // x-tiles, y-tiles, N*C
    gauss_blur_wmma<<<grid, dim3(256), 0, stream>>>(x, out);
}